// OuterProduct_22067541966983
// MI455X (gfx1250) — compile-verified
//
#include <hip/hip_runtime.h>
#include <hip/hip_bf16.h>
#include <stddef.h>

// Problem constants (from reference): BSZ=1, N=384, D_ATOM=256, D_PAIR=128, D_HID=32
#define NN      384
#define D_ATOM  256
#define D_PAIR  128
#define D_HID   32
#define NCOL    (NN * D_PAIR)        // 49152 columns of the big GEMM
#define N_TILES (NN / 16)            // 24
#define C_TILES (NCOL / 16)          // 3072
#define TOT_WAVES (N_TILES * C_TILES) // 73728

typedef __attribute__((ext_vector_type(2))) float v2f;
typedef __attribute__((ext_vector_type(8))) float v8f;

// ---------------------------------------------------------------------------
// Kernel 1: ab = (m @ W_in^T + b_in) * op_mask ; split into a, b
// grid: 384 blocks x 64 threads (one thread per output feature h)
// ---------------------------------------------------------------------------
__global__ void opm_linear_in(const float* __restrict__ M,      // [384][256]
                              const float* __restrict__ W_in,   // [64][256]
                              const float* __restrict__ b_in,   // [64]
                              const float* __restrict__ op_mask,// [384]
                              float* __restrict__ Aout,         // [384][32]
                              float* __restrict__ Bout)         // [384][32]
{
    const int n = blockIdx.x;
    const int h = threadIdx.x;
    const float* mr = M + n * D_ATOM;
    const float* wr = W_in + h * D_ATOM;
    float s = 0.f;
#pragma unroll 8
    for (int d = 0; d < D_ATOM; ++d) s = fmaf(mr[d], wr[d], s);
    s = (s + b_in[h]) * op_mask[n];
    if (h < D_HID) Aout[n * D_HID + h] = s;
    else           Bout[n * D_HID + (h - D_HID)] = s;
}

// ---------------------------------------------------------------------------
// Kernel 2: t[(m*128+p)][i] = sum_j b[m][j] * W_out[p][i*32+j]
// grid: 384 blocks (one per m-row) x 256 threads, 16 (p,i) pairs per thread
// ---------------------------------------------------------------------------
__global__ void opm_make_t(const float* __restrict__ Bmat,   // [384][32]
                           const float* __restrict__ W_out,  // [128][1024]
                           float* __restrict__ T)            // [49152][32]
{
    __shared__ float bsh[D_HID];
    const int mrow = blockIdx.x;
    const int t = threadIdx.x;
    if (t < D_HID) bsh[t] = Bmat[mrow * D_HID + t];
    __syncthreads();
#pragma unroll
    for (int it = 0; it < 16; ++it) {
        const int pi = it * 256 + t;       // 0..4095 = p*32 + i
        const int p = pi >> 5;
        const int i = pi & 31;
        const float* w = W_out + p * (D_HID * D_HID) + i * D_HID;
        float s = 0.f;
#pragma unroll
        for (int j = 0; j < D_HID; ++j) s = fmaf(bsh[j], w[j], s);
        T[(size_t)(mrow * D_PAIR + p) * D_HID + i] = s;
    }
}

// ---------------------------------------------------------------------------
// Kernel 3: z[n][m][p] = (sum_i a[n][i]*t[m*128+p][i] + b_out[p])
//                        * op_norm * op_mask[n] * op_mask[m]
// GEMM D[384,49152] = A[384,32] x T[49152,32]^T via V_WMMA_F32_16X16X4_F32.
// One wave per 16x16 tile; 8 chained K=4 WMMA accumulations cover K=32.
// Exact grid -> EXEC all ones (WMMA requirement), no bounds branches.
// ---------------------------------------------------------------------------
__global__ void opm_pair_wmma(const float* __restrict__ A,       // [384][32]
                              const float* __restrict__ T,       // [49152][32]
                              const float* __restrict__ b_out,   // [128]
                              const float* __restrict__ op_mask, // [384]
                              const float* __restrict__ op_norm, // [1]
                              float* __restrict__ Z)             // [384*384*128]
{
    const int wave = (int)((blockIdx.x * blockDim.x + threadIdx.x) >> 5);
    const int lane = threadIdx.x & 31;

    // column tile fastest -> consecutive waves share the A tile (L2-friendly)
    const int n_tile = wave / C_TILES;     // 0..23
    const int c_tile = wave % C_TILES;     // 0..3071
    const int n0 = n_tile * 16;
    const int c0 = c_tile * 16;            // global column = m*128 + p
    const int m  = c0 >> 7;                // fixed across the 16-wide tile
    const int p0 = c0 & (D_PAIR - 1);

    const int half = lane >> 4;            // 0: K={0,1}+k0, 1: K={2,3}+k0
    const int l15  = lane & 15;
    const int kb   = half * 2;

    // A fragment: lane covers row (l15), k = k0 + kb + {0,1}  (contiguous pair)
    const float* Arow = A + (size_t)(n0 + l15) * D_HID + kb;
    // B fragment = T^T: lane covers column (l15) -> T row (c0 + l15)
    const float* Trow = T + (size_t)(c0 + l15) * D_HID + kb;

    v8f acc = {};
#pragma unroll
    for (int k0 = 0; k0 < D_HID; k0 += 4) {
        const v2f af = *(const v2f*)(Arow + k0);
        const v2f bf = *(const v2f*)(Trow + k0);
        // (neg_a, A, neg_b, B, c_mod, C, reuse_a, reuse_b)
        acc = __builtin_amdgcn_wmma_f32_16x16x4_f32(
                  false, af, false, bf, (short)0, acc, false, false);
    }

    // Epilogue: bias + scale + masks, then store.
    const float norm = op_norm[0];
    const float mask_m = op_mask[m];
    const int   pcol = p0 + l15;
    const float bias = b_out[pcol];
    float* zbase = Z + (size_t)m * D_PAIR + pcol;   // + n * (384*128)

    // D layout: lane l, VGPR r -> row (r + 8*half), col l15
#pragma unroll
    for (int r = 0; r < 8; ++r) {
        const int n = n0 + r + 8 * half;
        const float v = (acc[r] + bias) * norm * op_mask[n] * mask_m;
        zbase[(size_t)n * NCOL] = v;
    }
}

// ---------------------------------------------------------------------------
extern "C" void kernel_launch(void* const* d_in, const int* in_sizes, int n_in,
                              void* d_out, int out_size, void* d_ws, size_t ws_size,
                              hipStream_t stream) {
    // setup_inputs() order: m, op_mask, op_norm, W_in, b_in, W_out, b_out
    const float* m       = (const float*)d_in[0];
    const float* op_mask = (const float*)d_in[1];
    const float* op_norm = (const float*)d_in[2];
    const float* W_in    = (const float*)d_in[3];
    const float* b_in    = (const float*)d_in[4];
    const float* W_out   = (const float*)d_in[5];
    const float* b_out   = (const float*)d_in[6];
    float* Z = (float*)d_out;

    // Workspace layout (floats): A[384*32] | B[384*32] | T[49152*32]  (~6.1 MB)
    float* Aw = (float*)d_ws;
    float* Bw = Aw + NN * D_HID;
    float* Tw = Bw + NN * D_HID;

    opm_linear_in<<<NN, 2 * D_HID, 0, stream>>>(m, W_in, b_in, op_mask, Aw, Bw);
    opm_make_t<<<NN, 256, 0, stream>>>(Bw, W_out, Tw);

    // 73728 waves, 8 waves (256 threads) per block -> 9216 blocks, exact cover.
    opm_pair_wmma<<<TOT_WAVES / 8, 256, 0, stream>>>(Aw, Tw, b_out, op_mask,
                                                     op_norm, Z);
}